// DistanceCorrelationLoss_31963146617373
// MI455X (gfx1250) — compile-verified
//
#include <hip/hip_runtime.h>
#include <hip/hip_bf16.h>
#include <math.h>

// ---------------------------------------------------------------------------
// Distance correlation loss, N=4096, DIM=512, fp32.
//   G = feat @ feat.T         (fp32 WMMA 16x16x4)
//   a = sqrt(max(sq_i + sq_j - 2G, 0)), diag = 0
//   A = a - row/(n-2) - col/(n-2) + tot/((n-1)(n-2)), diag = 0
//   result = <A,B> / max(sqrt(<A,A><B,B>), 1e-9)   (all /denom)
// ---------------------------------------------------------------------------

typedef __attribute__((ext_vector_type(2))) float v2f;
typedef __attribute__((ext_vector_type(8))) float v8f;

#define NPTS 4096
#define KDIM 512

// ---------------- small utility kernels ----------------

__global__ void zero_kernel(float* __restrict__ p, int n) {
    int i = blockIdx.x * blockDim.x + threadIdx.x;
    if (i < n) p[i] = 0.0f;
}

// Per-row squared norm: one wave per row, 8 rows per 256-thread block.
__global__ __launch_bounds__(256) void sq_kernel(const float* __restrict__ feat,
                                                 float* __restrict__ sq) {
    const int wave = threadIdx.x >> 5;
    const int lane = threadIdx.x & 31;
    const int row  = blockIdx.x * 8 + wave;
    const float* p = feat + (size_t)row * KDIM;
    float s = 0.0f;
#pragma unroll
    for (int k = lane; k < KDIM; k += 32) {
        float v = p[k];
        s = fmaf(v, v, s);
    }
#pragma unroll
    for (int off = 16; off > 0; off >>= 1) s += __shfl_down(s, off, 32);
    if (lane == 0) sq[row] = s;
}

// ---------------- WMMA distance-matrix GEMM ----------------
// Block = 256 threads = 8 waves. Block tile: 256 (M) x 64 (N).
// Each wave computes a 32 (M) x 64 (N) strip = 2x4 WMMA tiles, so per k-step:
//   2 A-fragment loads + 4 B-fragment loads feed 8 WMMAs (0.75 loads/WMMA).
//
// fp32 WMMA 16x16x4 fragment layouts (wave32):
//   A (16x4,  MxK): lanes 0-15 -> M=lane, holds K={0,1}; lanes 16-31 -> K={2,3}
//   B (4x16,  KxN): lanes 0-15 -> N=lane, holds K={0,1}; lanes 16-31 -> K={2,3}
//   C/D (16x16)   : VGPR v: lanes 0-15 -> M=v, lanes 16-31 -> M=v+8; N=lane%16
// Since B = feat^T, both A and B fragments are contiguous float2 loads of feat.
__global__ __launch_bounds__(256) void dist_gemm(const float* __restrict__ feat,
                                                 const float* __restrict__ sq,
                                                 float* __restrict__ a_out) {
    const int wave = threadIdx.x >> 5;
    const int lane = threadIdx.x & 31;
    const int half = lane >> 4;   // 0: K pair {0,1}, 1: K pair {2,3}
    const int l16  = lane & 15;

    const int m0 = blockIdx.y * 256 + wave * 32;
    const int n0 = blockIdx.x * 64;

    const float* Ap0 = feat + (size_t)(m0 + l16) * KDIM + 2 * half;
    const float* Ap1 = Ap0 + (size_t)16 * KDIM;
    const float* Bp  = feat + (size_t)(n0 + l16) * KDIM + 2 * half;

    v8f acc[2][4];
#pragma unroll
    for (int mi = 0; mi < 2; ++mi)
#pragma unroll
        for (int j = 0; j < 4; ++j) acc[mi][j] = v8f{};

#pragma unroll 2
    for (int k = 0; k < KDIM; k += 4) {
        v2f a0 = *(const v2f*)(Ap0 + k);
        v2f a1 = *(const v2f*)(Ap1 + k);
#pragma unroll
        for (int j = 0; j < 4; ++j) {
            v2f b = *(const v2f*)(Bp + (size_t)j * 16 * KDIM + k);
            acc[0][j] = __builtin_amdgcn_wmma_f32_16x16x4_f32(
                false, a0, false, b, (short)0, acc[0][j], false, false);
            acc[1][j] = __builtin_amdgcn_wmma_f32_16x16x4_f32(
                false, a1, false, b, (short)0, acc[1][j], false, false);
        }
    }

    // Epilogue: G -> Euclidean distance, write tiles to workspace.
    float sn[4];
#pragma unroll
    for (int j = 0; j < 4; ++j) sn[j] = sq[n0 + j * 16 + l16];

#pragma unroll
    for (int mi = 0; mi < 2; ++mi) {
        const int mbase = m0 + mi * 16 + half * 8;
        float sm[8];
#pragma unroll
        for (int v = 0; v < 8; ++v) sm[v] = sq[mbase + v];
#pragma unroll
        for (int j = 0; j < 4; ++j) {
            const int cn = n0 + j * 16 + l16;
#pragma unroll
            for (int v = 0; v < 8; ++v) {
                const int rm = mbase + v;
                float d2 = sm[v] + sn[j] - 2.0f * acc[mi][j][v];
                float d  = (d2 > 0.0f) ? sqrtf(d2) : 0.0f;
                if (rm == cn) d = 0.0f;   // exact zero diagonal (matches reference)
                a_out[(size_t)rm * NPTS + cn] = d;
            }
        }
    }
}

// ---------------- row sums + total ----------------
__global__ __launch_bounds__(256) void rowsum_kernel(const float* __restrict__ a,
                                                     float* __restrict__ row,
                                                     float* __restrict__ tot) {
    __shared__ float red[256];
    const int i = blockIdx.x;
    const float* p = a + (size_t)i * NPTS;
    float s = 0.0f;
    for (int j = threadIdx.x; j < NPTS; j += 256) s += p[j];
    red[threadIdx.x] = s;
    __syncthreads();
#pragma unroll
    for (int off = 128; off > 0; off >>= 1) {
        if (threadIdx.x < (unsigned)off) red[threadIdx.x] += red[threadIdx.x + off];
        __syncthreads();
    }
    if (threadIdx.x == 0) {
        row[i] = red[0];
        atomicAdd(tot, red[0]);
    }
}

// ---------------- centered dot products ----------------
__global__ __launch_bounds__(256) void dot_kernel(const float* __restrict__ aA,
                                                  const float* __restrict__ aB,
                                                  const float* __restrict__ rowA,
                                                  const float* __restrict__ rowB,
                                                  const float* __restrict__ scal,  // [totA, totB]
                                                  float* __restrict__ acc3) {
    const float inv = 1.0f / (float)(NPTS - 2);
    const float cA  = scal[0] / ((float)(NPTS - 1) * (float)(NPTS - 2));
    const float cB  = scal[1] / ((float)(NPTS - 1) * (float)(NPTS - 2));

    float sab = 0.0f, saa = 0.0f, sbb = 0.0f;
    const size_t total  = (size_t)NPTS * NPTS;
    const size_t stride = (size_t)gridDim.x * blockDim.x;
    for (size_t idx = (size_t)blockIdx.x * blockDim.x + threadIdx.x; idx < total;
         idx += stride) {
        const int i = (int)(idx >> 12);
        const int j = (int)(idx & (NPTS - 1));
        float Aij = aA[idx] - (rowA[i] + rowA[j]) * inv + cA;
        float Bij = aB[idx] - (rowB[i] + rowB[j]) * inv + cB;
        if (i == j) { Aij = 0.0f; Bij = 0.0f; }
        sab = fmaf(Aij, Bij, sab);
        saa = fmaf(Aij, Aij, saa);
        sbb = fmaf(Bij, Bij, sbb);
    }

    __shared__ float r0[256], r1[256], r2[256];
    const int tid = threadIdx.x;
    r0[tid] = sab; r1[tid] = saa; r2[tid] = sbb;
    __syncthreads();
#pragma unroll
    for (int off = 128; off > 0; off >>= 1) {
        if (tid < off) {
            r0[tid] += r0[tid + off];
            r1[tid] += r1[tid + off];
            r2[tid] += r2[tid + off];
        }
        __syncthreads();
    }
    if (tid == 0) {
        atomicAdd(&acc3[0], r0[0]);
        atomicAdd(&acc3[1], r1[0]);
        atomicAdd(&acc3[2], r2[0]);
    }
}

__global__ void final_kernel(const float* __restrict__ acc3, float* __restrict__ out) {
    const float denom = (float)NPTS * (float)(NPTS - 3);
    const float ab = acc3[0] / denom;
    const float aa = acc3[1] / denom;
    const float bb = acc3[2] / denom;
    float d = sqrtf(aa * bb);
    d = fmaxf(d, 1e-9f);
    out[0] = ab / d;
}

// ---------------- host launcher ----------------

extern "C" void kernel_launch(void* const* d_in, const int* in_sizes, int n_in,
                              void* d_out, int out_size, void* d_ws, size_t ws_size,
                              hipStream_t stream) {
    (void)in_sizes; (void)n_in; (void)out_size; (void)ws_size;
    const float* fa = (const float*)d_in[0];
    const float* fb = (const float*)d_in[1];
    float* out = (float*)d_out;

    // Workspace layout (needs ~134.25 MB):
    //   aA : NPTS*NPTS floats (64 MB)
    //   aB : NPTS*NPTS floats (64 MB)
    //   sqA, sqB, rowA, rowB : NPTS floats each
    //   scal[2] = {totA, totB}; acc3[3] = {<A,B>, <A,A>, <B,B>}
    const size_t mat = (size_t)NPTS * NPTS;
    char* ws = (char*)d_ws;
    float* aA   = (float*)ws;
    float* aB   = aA + mat;
    float* sqA  = aB + mat;
    float* sqB  = sqA + NPTS;
    float* rowA = sqB + NPTS;
    float* rowB = rowA + NPTS;
    float* scal = rowB + NPTS;   // 2 floats
    float* acc3 = scal + 2;      // 3 floats

    zero_kernel<<<1, 32, 0, stream>>>(scal, 5);

    sq_kernel<<<NPTS / 8, 256, 0, stream>>>(fa, sqA);
    sq_kernel<<<NPTS / 8, 256, 0, stream>>>(fb, sqB);

    dim3 grid(NPTS / 64, NPTS / 256);  // 64 x 16 blocks, 8 waves each
    dist_gemm<<<grid, 256, 0, stream>>>(fa, sqA, aA);
    dist_gemm<<<grid, 256, 0, stream>>>(fb, sqB, aB);

    rowsum_kernel<<<NPTS, 256, 0, stream>>>(aA, rowA, &scal[0]);
    rowsum_kernel<<<NPTS, 256, 0, stream>>>(aB, rowB, &scal[1]);

    dot_kernel<<<2048, 256, 0, stream>>>(aA, aB, rowA, rowB, scal, acc3);

    final_kernel<<<1, 1, 0, stream>>>(acc3, out);
}